// Relational_Attention_Layer_69226282877399
// MI455X (gfx1250) — compile-verified
//
#include <hip/hip_runtime.h>

// Relational attention layer, MI455X (gfx1250), fp32 via V_WMMA_F32_16X16X4_F32.
// B=8, N=512, D=256, KG=64. Dominant cost: one 512MB streaming pass over
// KG_matrix fused with a 64x64 fp32 WMMA GEMM + relu + weighted row-reduce.
// Tile staging uses the CDNA5 async global->LDS path (ASYNCcnt) to bypass VGPRs.

#define Dm 256
#define KGC 64
#define BB 8
#define NN 512
#define EPSV 1e-5f
#define SLOPE 0.01f   // jax.nn.leaky_relu default

typedef float v2f __attribute__((ext_vector_type(2)));
typedef float v4f __attribute__((ext_vector_type(4)));
typedef float v8f __attribute__((ext_vector_type(8)));

__device__ __forceinline__ v8f wmma4(v2f a, v2f b, v8f c) {
  // (neg_a, A, neg_b, B, c_mod, C, reuse_a, reuse_b)
  return __builtin_amdgcn_wmma_f32_16x16x4_f32(false, a, false, b, (short)0, c,
                                               false, false);
}

// CDNA5 async global->LDS copy, 16 bytes per lane. LDS operand = low 32 bits
// of the generic shared pointer (ISA: generic addr[31:0] is the wave-relative
// LDS byte address). Tracked with ASYNCcnt.
__device__ __forceinline__ void async_ld_b128(float* lds, const float* gptr) {
  unsigned loff = (unsigned)(unsigned long long)lds;
  asm volatile("global_load_async_to_lds_b128 %0, %1, off"
               :: "v"(loff), "v"(gptr)
               : "memory");
}
__device__ __forceinline__ void wait_async0() {
  asm volatile("s_wait_asynccnt 0" ::: "memory");
}

// ---------------------------------------------------------------------------
// Generic fused GEMM: out[M x 256] = act(A[M x KDIM] @ W[KDIM x 256] + bias)
//                                    (+ res). Block = 128 thr (4 waves), each
// wave owns a 16x16 output tile; 16-row A strip staged in LDS (row pad +4
// floats so half-wave ds_load_b64 A-fragment reads hit 64 distinct banks).
// BATCHW: W is per-batch (v matrix for attn @ v), batch = row/512.
// ---------------------------------------------------------------------------
template <int KDIM, int ACT, bool RES, bool BATCHW>
__global__ __launch_bounds__(128) void wmma_gemm(
    const float* __restrict__ A, const float* __restrict__ W,
    const float* __restrict__ bias, const float* __restrict__ res,
    float* __restrict__ out) {
  constexpr int LDA = KDIM + 4;
  __shared__ float sA[16 * LDA];
  const int tid  = threadIdx.x;
  const int wave = tid >> 5;
  const int lane = tid & 31;
  const int ln   = lane & 15;
  const int hi   = lane >> 4;      // 0 or 1 (half-wave)
  const int h2   = hi << 1;        // K sub-offset 0 or 2 (WMMA f32 A/B layout)
  const int rowG0 = blockIdx.x * 16;

  // Stage 16 x KDIM A strip: async global->LDS b128, no VGPR round-trip.
  constexpr int ITERS = (16 * KDIM / 4) / 128;
#pragma unroll
  for (int it = 0; it < ITERS; ++it) {
    int idx = it * 128 + tid;
    int r   = idx / (KDIM / 4);
    int c4  = idx % (KDIM / 4);
    async_ld_b128(sA + r * LDA + c4 * 4,
                  A + (size_t)(rowG0 + r) * KDIM + (size_t)c4 * 4);
  }
  wait_async0();
  __syncthreads();

  const float* Wb = W;
  if (BATCHW) Wb += (size_t)(rowG0 >> 9) * (size_t)NN * Dm;
  const int col = (blockIdx.y * 4 + wave) * 16 + ln;

  v8f acc = {};
#pragma unroll 8
  for (int ks = 0; ks < KDIM / 4; ++ks) {
    v2f a = *(const v2f*)(sA + ln * LDA + 4 * ks + h2);
    v2f b;
    b.x = Wb[(size_t)(4 * ks + h2) * Dm + col];
    b.y = Wb[(size_t)(4 * ks + h2 + 1) * Dm + col];
    acc = wmma4(a, b, acc);
  }

  const float bc = bias ? bias[col] : 0.0f;
#pragma unroll
  for (int e = 0; e < 8; ++e) {
    int row   = rowG0 + e + (hi << 3);   // C/D layout: VGPR e -> M=e (+8 hi half)
    float val = acc[e] + bc;
    if (ACT == 1) val = fmaxf(val, 0.0f);
    if (ACT == 2) val = (val > 0.0f) ? val : val * SLOPE;
    if (RES) val += res[(size_t)row * Dm + col];
    out[(size_t)row * Dm + col] = val;
  }
}

// ---------------------------------------------------------------------------
// Dominant kernel: skg[b,i,j] = wkg_p . relu(KG[b,i,j,:] @ Wkg + bkg)
// One block = 128 j-rows of one (b,i). 32KB tile async-staged to LDS once
// (the only read of KG_matrix). 4 waves x 2 j-tiles; Wkg B-fragments
// register-cached per 16-col group; bias+relu+weighted sum fused.
// ---------------------------------------------------------------------------
__global__ __launch_bounds__(128) void kg_kernel(
    const float* __restrict__ KG, const float* __restrict__ Wkg,
    const float* __restrict__ bkg, const float* __restrict__ wkgp,
    float* __restrict__ skg) {
  constexpr int LDK = KGC + 4;  // 68-float rows: conflict-free WMMA reads
  __shared__ float sA[128 * LDK];
  const int tid = threadIdx.x, wave = tid >> 5, lane = tid & 31;
  const int ln = lane & 15, hi = lane >> 4, h2 = hi << 1;
  const int blk  = blockIdx.x;
  const int jblk = blk & 3;           // 4 blocks of 128 rows per (b,i)
  const int bi   = blk >> 2;
  const size_t base = ((size_t)bi * NN + (size_t)jblk * 128) * KGC;

#pragma unroll
  for (int it = 0; it < 16; ++it) {
    int idx = it * 128 + tid;
    int r = idx >> 4, c4 = idx & 15;
    async_ld_b128(sA + r * LDK + c4 * 4,
                  KG + base + (size_t)r * KGC + (size_t)c4 * 4);
  }
  wait_async0();
  __syncthreads();

  float rs[2][8];
#pragma unroll
  for (int t = 0; t < 2; ++t)
#pragma unroll
    for (int e = 0; e < 8; ++e) rs[t][e] = 0.0f;

#pragma unroll
  for (int nt = 0; nt < 4; ++nt) {
    const int col = nt * 16 + ln;
    v2f bf[16];                          // Wkg fragments, register cached
#pragma unroll
    for (int ks = 0; ks < 16; ++ks) {
      bf[ks].x = Wkg[(4 * ks + h2) * KGC + col];
      bf[ks].y = Wkg[(4 * ks + h2 + 1) * KGC + col];
    }
    const float bc = bkg[col], wc = wkgp[col];
#pragma unroll
    for (int t = 0; t < 2; ++t) {
      const int jt = wave * 2 + t;
      v8f acc = {};
#pragma unroll
      for (int ks = 0; ks < 16; ++ks) {
        v2f a = *(const v2f*)(sA + (jt * 16 + ln) * LDK + 4 * ks + h2);
        acc = wmma4(a, bf[ks], acc);
      }
#pragma unroll
      for (int e = 0; e < 8; ++e) {
        float v = fmaxf(acc[e] + bc, 0.0f);  // relu(kg) then weight
        rs[t][e] += v * wc;
      }
    }
  }

  // Row-sum over N: reduce across the 16 lanes of each half-wave.
#pragma unroll
  for (int t = 0; t < 2; ++t)
#pragma unroll
    for (int e = 0; e < 8; ++e) {
      float v = rs[t][e];
      v += __shfl_xor(v, 1, 32);
      v += __shfl_xor(v, 2, 32);
      v += __shfl_xor(v, 4, 32);
      v += __shfl_xor(v, 8, 32);
      rs[t][e] = v;
    }
  if (ln == 0) {
#pragma unroll
    for (int t = 0; t < 2; ++t)
#pragma unroll
      for (int e = 0; e < 8; ++e) {
        int j = jblk * 128 + (wave * 2 + t) * 16 + e + (hi << 3);
        skg[(size_t)bi * NN + j] = rs[t][e];
      }
  }
}

// sq[row] = q[row,:].fc_w[0:256], sk[row] = k[row,:].fc_w[256:512]
__global__ __launch_bounds__(256) void sqsk_kernel(
    const float* __restrict__ q, const float* __restrict__ k,
    const float* __restrict__ fcw, float* __restrict__ sq,
    float* __restrict__ sk) {
  __shared__ float r0[256], r1[256];
  const int row = blockIdx.x, t = threadIdx.x;
  r0[t] = q[(size_t)row * Dm + t] * fcw[t];
  r1[t] = k[(size_t)row * Dm + t] * fcw[Dm + t];
  __syncthreads();
  for (int s = 128; s > 0; s >>= 1) {
    if (t < s) { r0[t] += r0[t + s]; r1[t] += r1[t + s]; }
    __syncthreads();
  }
  if (t == 0) { sq[row] = r0[0]; sk[row] = r1[0]; }
}

// In-place: scores = softmax_j(leaky_relu(skg + sq[i] + sk[j] + fc_b))
__global__ __launch_bounds__(128) void softmax_kernel(
    float* __restrict__ scores, const float* __restrict__ sq,
    const float* __restrict__ sk, const float* __restrict__ fcb) {
  __shared__ float red[128];
  const int bi = blockIdx.x, t = threadIdx.x;
  const int b = bi >> 9;
  const float fb = fcb[0], sqi = sq[bi];
  float x[4], mx = -1e30f;
#pragma unroll
  for (int u = 0; u < 4; ++u) {
    int j = u * 128 + t;
    float s = scores[(size_t)bi * NN + j] + sqi + sk[(size_t)b * NN + j] + fb;
    s = (s > 0.0f) ? s : s * SLOPE;
    x[u] = s;
    mx = fmaxf(mx, s);
  }
  red[t] = mx; __syncthreads();
  for (int s = 64; s > 0; s >>= 1) {
    if (t < s) red[t] = fmaxf(red[t], red[t + s]);
    __syncthreads();
  }
  mx = red[0]; __syncthreads();
  float sum = 0.0f;
#pragma unroll
  for (int u = 0; u < 4; ++u) { x[u] = __expf(x[u] - mx); sum += x[u]; }
  red[t] = sum; __syncthreads();
  for (int s = 64; s > 0; s >>= 1) {
    if (t < s) red[t] += red[t + s];
    __syncthreads();
  }
  const float inv = 1.0f / red[0];
#pragma unroll
  for (int u = 0; u < 4; ++u)
    scores[(size_t)bi * NN + u * 128 + t] = x[u] * inv;
}

// LayerNorm over D=256 per row.
__global__ __launch_bounds__(256) void ln_kernel(
    const float* __restrict__ src, const float* __restrict__ g,
    const float* __restrict__ bta, float* __restrict__ out) {
  __shared__ float r0[256], r1[256];
  const int row = blockIdx.x, t = threadIdx.x;
  const float x = src[(size_t)row * Dm + t];
  r0[t] = x; r1[t] = x * x;
  __syncthreads();
  for (int s = 128; s > 0; s >>= 1) {
    if (t < s) { r0[t] += r0[t + s]; r1[t] += r1[t + s]; }
    __syncthreads();
  }
  const float mu  = r0[0] * (1.0f / Dm);
  const float var = r1[0] * (1.0f / Dm) - mu * mu;
  out[(size_t)row * Dm + t] = (x - mu) * rsqrtf(var + EPSV) * g[t] + bta[t];
}

extern "C" void kernel_launch(void* const* d_in, const int* in_sizes, int n_in,
                              void* d_out, int out_size, void* d_ws,
                              size_t ws_size, hipStream_t stream) {
  (void)in_sizes; (void)n_in; (void)out_size; (void)ws_size;
  const float* vectors = (const float*)d_in[0];
  const float* KGm = (const float*)d_in[1];
  const float* Wq = (const float*)d_in[2];  const float* bq = (const float*)d_in[3];
  const float* Wk = (const float*)d_in[4];  const float* bk = (const float*)d_in[5];
  const float* Wv = (const float*)d_in[6];  const float* bv = (const float*)d_in[7];
  const float* Wkg = (const float*)d_in[8]; const float* bkg = (const float*)d_in[9];
  const float* fcw = (const float*)d_in[10]; const float* fcb = (const float*)d_in[11];
  const float* lng = (const float*)d_in[12]; const float* lnb = (const float*)d_in[13];
  const float* W1 = (const float*)d_in[14]; const float* b1 = (const float*)d_in[15];
  const float* W2 = (const float*)d_in[16]; const float* b2 = (const float*)d_in[17];
  // d_in[18]/d_in[19] (Wo, bo) are unused by the reference.

  float* ws = (float*)d_ws;
  const size_t R = (size_t)BB * NN * Dm;  // 1,048,576
  float* q      = ws;
  float* k      = q + R;
  float* v      = k + R;
  float* sq     = v + R;
  float* sk     = sq + (size_t)BB * NN;
  float* scores = sk + (size_t)BB * NN;
  float* src    = scores + (size_t)BB * NN * NN;
  float* src2   = src + R;
  float* hbuf   = src2 + R;
  float* outf   = (float*)d_out;

  const dim3 gg(BB * NN / 16, 4), gb(128);

  // q, k, v projections (relu)
  wmma_gemm<256, 1, false, false><<<gg, gb, 0, stream>>>(vectors, Wq, bq, nullptr, q);
  wmma_gemm<256, 1, false, false><<<gg, gb, 0, stream>>>(vectors, Wk, bk, nullptr, k);
  wmma_gemm<256, 1, false, false><<<gg, gb, 0, stream>>>(vectors, Wv, bv, nullptr, v);
  // sq, sk scalars
  sqsk_kernel<<<BB * NN, 256, 0, stream>>>(q, k, fcw, sq, sk);
  // fused kg GEMM + relu + weighted reduce  (the 512MB streaming pass)
  kg_kernel<<<BB * NN * 4, 128, 0, stream>>>(KGm, Wkg, bkg, fcw + 2 * Dm, scores);
  // leaky_relu + softmax over j (in place)
  softmax_kernel<<<BB * NN, 128, 0, stream>>>(scores, sq, sk, fcb);
  // src = v + attn_weights @ v   (batched W = v[b])
  wmma_gemm<512, 0, true, true><<<gg, gb, 0, stream>>>(scores, v, nullptr, v, src);
  // LayerNorm
  ln_kernel<<<BB * NN, 256, 0, stream>>>(src, lng, lnb, src2);
  // FFN: h = leaky(src2 @ W1 + b1); out = src + h @ W2 + b2
  wmma_gemm<256, 2, false, false><<<gg, gb, 0, stream>>>(src2, W1, b1, nullptr, hbuf);
  wmma_gemm<256, 0, true, false><<<gg, gb, 0, stream>>>(hbuf, W2, b2, src, outf);
}